// TMSCNNAveragePooling_74294344286543
// MI455X (gfx1250) — compile-verified
//
#include <hip/hip_runtime.h>
#include <hip/hip_bf16.h>

#define CCH 64                 // channels
#define TILE_E 64              // elements per tile
#define THREADS 256            // 8 wave32
#define TILE_VALS (TILE_E * CCH)      // 4096 floats = 16 KB

// Low 32 bits of a generic pointer to an LDS object are the LDS byte offset
// (flat LDS aperture: addr[31:0] = offset; see CDNA5 ISA aperture rules).
__device__ __forceinline__ unsigned lds_off(const void* p) {
  return (unsigned)(uintptr_t)p;
}

__global__ void seg_zero_kernel(float4* out4, long n_out4, float4* cnt4, long n_cnt4) {
  long i = (long)blockIdx.x * blockDim.x + threadIdx.x;
  float4 z = make_float4(0.f, 0.f, 0.f, 0.f);
  if (i < n_out4) {
    out4[i] = z;
  } else if (i - n_out4 < n_cnt4) {
    cnt4[i - n_out4] = z;
  }
}

__global__ __launch_bounds__(THREADS)
void seg_scatter_kernel(const int* __restrict__ mesh_ids,
                        const float* __restrict__ src,
                        float* __restrict__ sums,
                        float* __restrict__ counts,
                        int num_tiles) {
  __shared__ float buf[2][TILE_VALS];   // 2 x 16 KB double buffer
  __shared__ int   ids[2][TILE_E];

  const int t = threadIdx.x;

  // Stage one 64-element x 64-channel tile into LDS via the async data path.
  // Each lane moves 64 B (4 x b128); per wave that is 4 async instructions
  // (ASYNCcnt += 4). INST_OFFSET advances both the global and LDS address.
  // TH_LOAD_NT: the 512 MB source stream has zero reuse — keep it from
  // evicting the 130 MB L2-resident accumulator region the atomics hit.
  auto stage = [&](int tile, int b) {
    uint64_t ga = (uint64_t)(uintptr_t)(src + (size_t)tile * TILE_VALS)
                + (unsigned)(t * 16);
    unsigned la = lds_off(&buf[b][0]) + (unsigned)(t * 16);
    asm volatile(
        "global_load_async_to_lds_b128 %0, %1, off th:TH_LOAD_NT\n\t"
        "global_load_async_to_lds_b128 %0, %1, off offset:4096 th:TH_LOAD_NT\n\t"
        "global_load_async_to_lds_b128 %0, %1, off offset:8192 th:TH_LOAD_NT\n\t"
        "global_load_async_to_lds_b128 %0, %1, off offset:12288 th:TH_LOAD_NT"
        :: "v"(la), "v"(ga) : "memory");
    if (t < TILE_E) ids[b][t] = mesh_ids[(size_t)tile * TILE_E + t];
  };

  int tile = blockIdx.x;               // uniform across block
  if (tile >= num_tiles) return;

  int b = 0;
  stage(tile, 0);

  for (; tile < num_tiles; tile += (int)gridDim.x) {
    int nxt = tile + (int)gridDim.x;
    if (nxt < num_tiles) {
      stage(nxt, b ^ 1);                                // 4 more in flight
      asm volatile("s_wait_asynccnt 0x4" ::: "memory"); // current tile done
    } else {
      asm volatile("s_wait_asynccnt 0x0" ::: "memory");
    }
    __syncthreads();   // publish this wave's LDS tile to all 8 waves

    // 64 lanes per element row: lane c -> channel c. Coalesced LDS reads and
    // coalesced 256B atomic bursts into the L2-resident accumulator.
    #pragma unroll
    for (int sub = 0; sub < TILE_VALS / THREADS; ++sub) {
      int idx = t + sub * THREADS;     // 0..4095
      int e   = idx >> 6;              // element within tile
      int c   = idx & (CCH - 1);       // channel
      int m   = ids[b][e];
      float v = buf[b][idx];
      unsafeAtomicAdd(&sums[(size_t)m * CCH + c], v);   // native f32 fadd, no-return
      if (c == 0) unsafeAtomicAdd(&counts[m], 1.0f);
    }
    __syncthreads();   // all waves done reading buf[b] before it is restaged
    b ^= 1;
  }
}

__global__ void seg_normalize_kernel(float* __restrict__ out,
                                     const float* __restrict__ counts,
                                     long n4) {
  long i = (long)blockIdx.x * blockDim.x + threadIdx.x;
  if (i >= n4) return;
  int m = (int)(i >> 4);               // 16 float4 per 64-channel row
  float s = 1.0f / fmaxf(counts[m], 1.0f);
  float4 v = ((const float4*)out)[i];
  v.x *= s; v.y *= s; v.z *= s; v.w *= s;
  ((float4*)out)[i] = v;
}

extern "C" void kernel_launch(void* const* d_in, const int* in_sizes, int n_in,
                              void* d_out, int out_size, void* d_ws, size_t ws_size,
                              hipStream_t stream) {
  const int*   mesh_ids = (const int*)d_in[0];
  const float* sources  = (const float*)d_in[1];
  // d_in[2] is the num_segments scalar; M is recovered from out_size instead.
  const int N = in_sizes[0];                 // 2,097,152
  const int M = out_size / CCH;              // 524,288

  float* out    = (float*)d_out;             // [M, 64] accumulators -> final means
  float* counts = (float*)d_ws;              // [M] f32 counts (2 MB of scratch)

  // 1) zero accumulators + counts (atomics need a clean base every call)
  long n_out4 = (long)M * CCH / 4;
  long n_cnt4 = (long)M / 4;
  long tot4   = n_out4 + n_cnt4;
  int  zblk   = (int)((tot4 + THREADS - 1) / THREADS);
  seg_zero_kernel<<<zblk, THREADS, 0, stream>>>((float4*)out, n_out4,
                                                (float4*)counts, n_cnt4);

  // 2) async-staged scatter-add (sums + counts), accumulators live in L2;
  //    source stream marked non-temporal so it cannot thrash them
  int num_tiles = N / TILE_E;                // 32768
  int blocks = 8192;                         // 4 tiles/block, deep pipelining
  if (blocks > num_tiles) blocks = num_tiles;
  seg_scatter_kernel<<<blocks, THREADS, 0, stream>>>(mesh_ids, sources,
                                                     out, counts, num_tiles);

  // 3) in-place mean (L2-hit reads, streamed writes)
  int nblk = (int)((n_out4 + THREADS - 1) / THREADS);
  seg_normalize_kernel<<<nblk, THREADS, 0, stream>>>(out, counts, n_out4);
}